// Attention_37168646979650
// MI455X (gfx1250) — compile-verified
//
#include <hip/hip_runtime.h>
#include <math.h>

// ---------------------------------------------------------------------------
// GAT-style attention for MI455X (gfx1250), f32 WMMA 16x16x4 throughout.
//   hp  = h @ W^T                       [8192,128]
//   a   = softmax(hp@hp^T - 1e10*I)     [8192,8192]  (diag -> 1 after +I,clamp)
//   out = a @ hp                        [8192,128]
// d_out = concat(out, a) flat f32.
// a-stream written with TH=NT (b128) to keep hp resident in L2.
// ---------------------------------------------------------------------------

typedef float v2f __attribute__((ext_vector_type(2)));
typedef float v4f __attribute__((ext_vector_type(4)));
typedef float v8f __attribute__((ext_vector_type(8)));

#define NROW   8192
#define IND    256
#define OUTD   128
#define TS     16              // WMMA tile
#define RT     (NROW / TS)     // 512 row tiles
#define CT     (NROW / TS)     // 512 col tiles
#define CSPLIT 8               // column chunks per row tile
#define CHUNK  (CT / CSPLIT)   // 64 col tiles per chunk

__device__ __forceinline__ v8f wmma4(v2f a, v2f b, v8f c) {
  // D = A(16x4 f32) * B(4x16 f32) + C(16x16 f32)
  return __builtin_amdgcn_wmma_f32_16x16x4_f32(
      /*neg_a=*/false, a, /*neg_b=*/false, b,
      /*c_mod=*/(short)0, c, /*reuse_a=*/false, /*reuse_b=*/false);
}

// ---------------- Kernel 1: hp = h @ W^T -----------------------------------
// One wave per 16x16 output tile. A lane: row = lane%16, K-half = lane/16.
__global__ void __launch_bounds__(256)
k_hp(const float* __restrict__ h, const float* __restrict__ W,
     float* __restrict__ hp) {
  const int lane = threadIdx.x;                  // 0..31
  const int tile = blockIdx.x * 8 + threadIdx.y; // 0..4095
  const int mt = tile >> 3;                      // 0..511
  const int nt = tile & 7;                       // 0..7
  const int r  = lane & 15;
  const int hi = lane >> 4;

  const float* arow = h + (size_t)(mt * TS + r) * IND + 2 * hi;  // A row r
  const float* brow = W + (size_t)(nt * TS + r) * IND + 2 * hi;  // B col r (= W row)

  v8f c = {};
#pragma unroll 8
  for (int k = 0; k < IND; k += 4) {
    v2f a = *(const v2f*)(arow + k);
    v2f b = *(const v2f*)(brow + k);
    c = wmma4(a, b, c);
  }
#pragma unroll
  for (int i = 0; i < 8; ++i)
    hp[(size_t)(mt * TS + i + 8 * hi) * OUTD + nt * TS + r] = c[i];
}

// ---------------- Kernel 2: per-(row tile, chunk) softmax stats ------------
__global__ void __launch_bounds__(256)
k_stats(const float* __restrict__ hp,
        float* __restrict__ partMax, float* __restrict__ partSum) {
  const int lane = threadIdx.x;
  const int wid  = blockIdx.x * 8 + threadIdx.y;   // 0..4095
  const int t  = wid / CSPLIT;                     // row tile
  const int cc = wid % CSPLIT;                     // column chunk
  const int r  = lane & 15;
  const int hi = lane >> 4;

  // Hoist the whole A strip (16 rows x 128 K) into 32 v2f registers.
  const float* arow = hp + (size_t)(t * TS + r) * OUTD + 2 * hi;
  v2f areg[OUTD / 4];
#pragma unroll
  for (int k = 0; k < OUTD / 4; ++k) areg[k] = *(const v2f*)(arow + 4 * k);

  float m[8], l[8];
#pragma unroll
  for (int i = 0; i < 8; ++i) { m[i] = -INFINITY; l[i] = 0.0f; }

  for (int j = cc * CHUNK; j < cc * CHUNK + CHUNK; ++j) {
    const float* brow = hp + (size_t)(j * TS + r) * OUTD + 2 * hi;
    if (j + 1 < cc * CHUNK + CHUNK)
      __builtin_prefetch(brow + OUTD * TS, 0, 1);  // next B tile -> global_prefetch
    v8f c = {};
#pragma unroll
    for (int k = 0; k < OUTD / 4; ++k) {
      v2f b = *(const v2f*)(brow + 4 * k);
      c = wmma4(areg[k], b, c);
    }
    const bool diagTile = (j == t);
#pragma unroll
    for (int i = 0; i < 8; ++i) {
      const bool isDiag = diagTile && ((i + 8 * hi) == r);
      float v = isDiag ? -INFINITY : c[i];
      float tm = v;                               // max over 16 cols (half-group)
      for (int s = 1; s < 16; s <<= 1) tm = fmaxf(tm, __shfl_xor(tm, s, 32));
      const float nm = fmaxf(m[i], tm);
      float e = isDiag ? 0.0f : __expf(v - nm);
      float ts = e;                               // sum over 16 cols
      for (int s = 1; s < 16; s <<= 1) ts += __shfl_xor(ts, s, 32);
      l[i] = l[i] * __expf(m[i] - nm) + ts;
      m[i] = nm;
    }
  }
  if (r == 0) {                                   // lanes 0 and 16 write 8 rows each
#pragma unroll
    for (int i = 0; i < 8; ++i) {
      const int row = t * TS + 8 * hi + i;
      partMax[(size_t)cc * NROW + row] = m[i];
      partSum[(size_t)cc * NROW + row] = l[i];
    }
  }
}

// ---------------- Kernel 3: combine chunk partials -------------------------
__global__ void k_combine(const float* __restrict__ partMax,
                          const float* __restrict__ partSum,
                          float* __restrict__ rowMax, float* __restrict__ rowSum) {
  const int row = blockIdx.x * blockDim.x + threadIdx.x;
  if (row >= NROW) return;
  float m = -INFINITY;
#pragma unroll
  for (int c = 0; c < CSPLIT; ++c) m = fmaxf(m, partMax[(size_t)c * NROW + row]);
  float s = 0.0f;
#pragma unroll
  for (int c = 0; c < CSPLIT; ++c)
    s += partSum[(size_t)c * NROW + row] * __expf(partMax[(size_t)c * NROW + row] - m);
  rowMax[row] = m;
  rowSum[row] = s;
}

// ---------------- Kernel 4: zero the `out` region --------------------------
__global__ void k_zero(float* __restrict__ p, int n) {
  const int i = blockIdx.x * blockDim.x + threadIdx.x;
  if (i < n) p[i] = 0.0f;
}

// ---------------- Kernel 5: write a (NT b128), accumulate out = a @ hp -----
__global__ void __launch_bounds__(256)
k_attend(const float* __restrict__ hp,
         const float* __restrict__ rowMax, const float* __restrict__ rowSum,
         float* __restrict__ a, float* __restrict__ out) {
  __shared__ float smem[8][TS * TS];              // per-wave 16x16 P staging
  const int lane = threadIdx.x;
  const int wave = threadIdx.y;
  const int wid  = blockIdx.x * 8 + wave;         // 0..4095
  const int t  = wid / CSPLIT;
  const int cc = wid % CSPLIT;
  const int r  = lane & 15;
  const int hi = lane >> 4;
  float* sp = smem[wave];

  float mrow[8], inv[8];
#pragma unroll
  for (int i = 0; i < 8; ++i) {
    const int row = t * TS + i + 8 * hi;
    mrow[i] = rowMax[row];
    inv[i]  = 1.0f / rowSum[row];
  }

  // Hoisted A strip for the score GEMM.
  const float* arow = hp + (size_t)(t * TS + r) * OUTD + 2 * hi;
  v2f areg[OUTD / 4];
#pragma unroll
  for (int k = 0; k < OUTD / 4; ++k) areg[k] = *(const v2f*)(arow + 4 * k);

  v8f acc[8];
#pragma unroll
  for (int n = 0; n < 8; ++n) acc[n] = (v8f){};

  // a-store lane mapping: lane -> (row = lane/2, 8-col half = lane&1)
  const int srow  = lane >> 1;
  const int shalf = lane & 1;

  for (int j = cc * CHUNK; j < cc * CHUNK + CHUNK; ++j) {
    // ---- scores tile: S = hp_rows @ hp_cols^T --------------------------
    const float* brow = hp + (size_t)(j * TS + r) * OUTD + 2 * hi;
    if (j + 1 < cc * CHUNK + CHUNK)
      __builtin_prefetch(brow + OUTD * TS, 0, 1);
    v8f c = {};
#pragma unroll
    for (int k = 0; k < OUTD / 4; ++k) {
      v2f bv = *(const v2f*)(brow + 4 * k);
      c = wmma4(areg[k], bv, c);
    }
    // ---- P = min(exp(S-m)/l, 1), diag -> 1; stage in LDS ---------------
    const bool diagTile = (j == t);
#pragma unroll
    for (int i = 0; i < 8; ++i) {
      const int rr = i + 8 * hi;                  // row within tile
      float p;
      if (diagTile && rr == r) p = 1.0f;          // min(softmax+I, 1) on diagonal
      else                     p = fminf(__expf(c[i] - mrow[i]) * inv[i], 1.0f);
      sp[rr * TS + r] = p;
    }
    // LDS is in-order per wave; fence the compiler + drain DScnt before the
    // cross-lane re-layout reads below.
    asm volatile("s_wait_dscnt 0" ::: "memory");

    // ---- write `a` tile: 2 x b128 per lane, non-temporal ----------------
    {
      const v4f* src = (const v4f*)(sp + srow * TS + shalf * 8);
      v4f* dst = (v4f*)(a + (size_t)(t * TS + srow) * NROW + (size_t)j * TS + shalf * 8);
      __builtin_nontemporal_store(src[0], dst);
      __builtin_nontemporal_store(src[1], dst + 1);
    }

    // ---- out += P @ hp_cols  (A operand re-read from LDS in A-layout) --
#pragma unroll
    for (int k4 = 0; k4 < 4; ++k4) {
      const int kk = 4 * k4 + 2 * hi;             // K index pair for this lane half
      v2f av = { sp[r * TS + kk], sp[r * TS + kk + 1] };
      const float* b0 = hp + (size_t)(j * TS + kk) * OUTD;
      const float* b1 = hp + (size_t)(j * TS + kk + 1) * OUTD;
#pragma unroll
      for (int n = 0; n < 8; ++n) {
        v2f bv = { b0[n * TS + r], b1[n * TS + r] };
        acc[n] = wmma4(av, bv, acc[n]);
      }
    }
  }

  // ---- commit partial out with f32 atomics --------------------------------
#pragma unroll
  for (int n = 0; n < 8; ++n)
#pragma unroll
    for (int i = 0; i < 8; ++i)
      atomicAdd(&out[(size_t)(t * TS + i + 8 * hi) * OUTD + n * TS + r], acc[n][i]);
}

// ---------------------------------------------------------------------------
extern "C" void kernel_launch(void* const* d_in, const int* in_sizes, int n_in,
                              void* d_out, int out_size, void* d_ws, size_t ws_size,
                              hipStream_t stream) {
  const float* h = (const float*)d_in[0];   // [8192,256]
  const float* W = (const float*)d_in[1];   // [128,256]

  float* out = (float*)d_out;                       // [8192,128]
  float* a   = out + (size_t)NROW * OUTD;           // [8192,8192]

  // workspace layout (floats): hp | partMax | partSum | rowMax | rowSum
  float* hp      = (float*)d_ws;                    // 1,048,576
  float* partMax = hp + (size_t)NROW * OUTD;        // 65,536
  float* partSum = partMax + (size_t)CSPLIT * NROW; // 65,536
  float* rowMax  = partSum + (size_t)CSPLIT * NROW; // 8,192
  float* rowSum  = rowMax + NROW;                   // 8,192

  const dim3 wblk(32, 8);

  // 1) hp = h @ W^T : 512*8 tiles, 8 waves/block
  k_hp<<<dim3(RT * (OUTD / TS) / 8), wblk, 0, stream>>>(h, W, hp);

  // 2) softmax stats over column chunks: 512*8 waves
  k_stats<<<dim3(RT * CSPLIT / 8), wblk, 0, stream>>>(hp, partMax, partSum);

  // 3) combine chunk partials per row
  k_combine<<<dim3((NROW + 255) / 256), dim3(256), 0, stream>>>(partMax, partSum,
                                                                rowMax, rowSum);

  // 4) zero `out` region (poisoned by harness) before atomic accumulation
  k_zero<<<dim3((NROW * OUTD + 255) / 256), dim3(256), 0, stream>>>(out, NROW * OUTD);

  // 5) write a (non-temporal), accumulate out
  k_attend<<<dim3(RT * CSPLIT / 8), wblk, 0, stream>>>(hp, rowMax, rowSum, a, out);
}